// netAttention_2370821947895
// MI455X (gfx1250) — compile-verified
//
#include <hip/hip_runtime.h>

#define NN 4096
#define DF 64

typedef __attribute__((ext_vector_type(2))) float v2f;
typedef __attribute__((ext_vector_type(8))) float v8f;

// ---------------------------------------------------------------------------
// Phase C (launched first): zero the dense att output region (mostly zeros;
// the scatter kernel fills in the <=13 nonzeros per row afterwards).
// ---------------------------------------------------------------------------
__global__ __launch_bounds__(256)
void zero_att(float4* __restrict__ att4) {
    size_t i = (size_t)blockIdx.x * blockDim.x + threadIdx.x;
    att4[i] = make_float4(0.f, 0.f, 0.f, 0.f);
}

// ---------------------------------------------------------------------------
// Phase A: denom[t] = sum_i exp(k_i . v_t) via f32 WMMA.
// One block = 4 waves, owns a 16-column tile of S = K @ V^T.
// Per ISA 7.12.2 (32-bit A 16x4): lane = m + 16*(kk>>1), vgpr = kk&1.
// B (4x16) mirrored with n in place of m. D (16x16 f32): lane n=L&15,
// rows m = 8*(L>>4) + r for VGPR r.
// ---------------------------------------------------------------------------
__global__ __launch_bounds__(128)
void col_sums(const float* __restrict__ Km, const float* __restrict__ V,
              float* __restrict__ denom) {
    __shared__ float red[4][16];
    const int lane  = threadIdx.x & 31;
    const int wv    = threadIdx.x >> 5;
    const int t0    = blockIdx.x * 16;
    const int n     = lane & 15;
    const int khalf = lane >> 4;  // selects K pair {0,1} vs {2,3} within a step

    // Preload B: 16 K-steps of the fixed 16-column v-tile (32 VGPRs).
    v2f b[16];
#pragma unroll
    for (int kk = 0; kk < 16; ++kk)
        b[kk] = *(const v2f*)(V + (size_t)(t0 + n) * DF + 4 * kk + 2 * khalf);

    float colsum = 0.f;
    for (int i0 = wv * 16; i0 < NN; i0 += 64) {
        v8f acc = {0.f, 0.f, 0.f, 0.f, 0.f, 0.f, 0.f, 0.f};
#pragma unroll
        for (int kk = 0; kk < 16; ++kk) {
            v2f a = *(const v2f*)(Km + (size_t)(i0 + n) * DF + 4 * kk + 2 * khalf);
            // D = A(16x4) * B(4x16) + C ; chained over K=64
            acc = __builtin_amdgcn_wmma_f32_16x16x4_f32(
                false, a, false, b[kk], (short)0, acc, false, false);
        }
#pragma unroll
        for (int r = 0; r < 8; ++r)
            colsum += __expf(acc[r]);
    }
    // Combine the two row-halves of each column (lane n with lane n+16).
    colsum += __shfl_xor(colsum, 16, 32);
    if (lane < 16) red[wv][lane] = colsum;
    __syncthreads();
    if (wv == 0 && lane < 16)
        denom[t0 + lane] = red[0][lane] + red[1][lane] + red[2][lane] + red[3][lane];
}

// ---------------------------------------------------------------------------
// Phase B: exploit coupon sparsity. One wave per column t walks the ancestor
// chain (<=13 nodes), writes the nonzero att entries and accumulates ans[t].
// Lanes split the 64 features (2 each).
// ---------------------------------------------------------------------------
__global__ __launch_bounds__(256)
void scatter(const float* __restrict__ Q, const float* __restrict__ Km,
             const float* __restrict__ V, const float* __restrict__ w,
             const int* __restrict__ parent, const float* __restrict__ denom,
             float* __restrict__ ans, float* __restrict__ att) {
    const int lane = threadIdx.x & 31;
    const int wv   = threadIdx.x >> 5;
    const int t    = blockIdx.x * 8 + wv;
    const int f    = lane * 2;

    const float vt0   = V[(size_t)t * DF + f];
    const float vt1   = V[(size_t)t * DF + f + 1];
    const float inv_d = 1.0f / denom[t];

    float a0 = 0.f, a1 = 0.f;
    int   cur = t;
    float c   = 1.0f;  // coupon product along the path s -> t

#pragma unroll 1
    for (int j = 0; j < 13; ++j) {
        const int s = cur;
        float p = Km[(size_t)s * DF + f] * vt0 + Km[(size_t)s * DF + f + 1] * vt1;
#pragma unroll
        for (int off = 16; off >= 1; off >>= 1)
            p += __shfl_xor(p, off, 32);
        const float val = __expf(p) * inv_d * c;  // softmax[s,t] * C[s,t]
        if (lane == 0) att[(size_t)t * NN + s] = val;
        a0 += val * Q[(size_t)s * DF + f];
        a1 += val * Q[(size_t)s * DF + f + 1];
        if (s == 0) break;          // reached root
        c *= w[cur - 1];            // w[node] = netParm[node-1]
        cur = parent[cur];
    }
    ans[(size_t)t * DF + f]     = a0;
    ans[(size_t)t * DF + f + 1] = a1;
}

// ---------------------------------------------------------------------------
extern "C" void kernel_launch(void* const* d_in, const int* in_sizes, int n_in,
                              void* d_out, int out_size, void* d_ws, size_t ws_size,
                              hipStream_t stream) {
    const float* q       = (const float*)d_in[0];
    const float* k       = (const float*)d_in[1];
    const float* v       = (const float*)d_in[2];
    const float* netParm = (const float*)d_in[3];
    const int*   parent  = (const int*)d_in[4];

    float* ans   = (float*)d_out;                     // [N, 64]
    float* att   = (float*)d_out + (size_t)NN * DF;   // [N, N], row-major [t, s]
    float* denom = (float*)d_ws;                      // [N]

    // 1) zero dense att (16.7M floats as float4)
    zero_att<<<(NN * (size_t)NN / 4) / 256, 256, 0, stream>>>((float4*)att);
    // 2) column softmax denominators via WMMA GEMM-reduction
    col_sums<<<NN / 16, 128, 0, stream>>>(k, v, denom);
    // 3) sparse coupon walk: att nonzeros + ans
    scatter<<<NN / 8, 256, 0, stream>>>(q, k, v, netParm, parent, denom, ans, att);
}